// Kalman_Smooth_Gradient_15135464751885
// MI455X (gfx1250) — compile-verified
//
#include <hip/hip_runtime.h>
#include <math.h>

// ---------------------------------------------------------------------------
// Kalman filter + RTS smoother + NLL eval, restructured for MI455X (gfx1250).
//
// Covariance/gain recursions are data-independent -> computed ONCE by a
// single wave using V_WMMA_F32_16X16X4_F32. All 6x6 matrices live in
// zero-padded 16x16 LDS tiles so WMMA operand gathers and result stores are
// unconditional (no exec-mask divergence). Per-trajectory work reduces to
// cheap batched state recursions.
// ---------------------------------------------------------------------------

#define BATCH   4096
#define TSTEPS  256
#define DT_     (1.0f / 120.0f)
#define P0S     0.01f

// workspace layout (float offsets)
#define K_OFF     0                      // 256*18
#define C_OFF     4608                   // 255*36
#define SINV_OFF  13788                  // 256*9
#define DETS_OFF  16092                  // 256
#define PART_OFF  16348                  // 16 block partials
#define XF_OFF    16384                  // [T][B][6]

typedef float v2f __attribute__((ext_vector_type(2)));
typedef float v8f __attribute__((ext_vector_type(8)));

__device__ __forceinline__ float wrapf(float a) {
    const float PI_  = 3.14159265358979323846f;
    const float TPI_ = 6.28318530717958647693f;
    return a - TPI_ * floorf((a + PI_) / TPI_);
}

// ---- WMMA helpers on zero-padded 16x16 tiles (stride 16) ------------------
// A-operand (32-bit A 16x4): lanes 0-15 M=lane, VGPR r = K=r; lanes 16-31
// K=2+r. B mirrored over N. C/D: VGPR r -> row M = r + 8*(lane>=16), N=lane%16.
// Padding rows/cols are zero, so the full 16x16 D equals the padded 6x6
// result: all loads/stores are unconditional.
template <bool BT>
__device__ __forceinline__ v8f mul6p(const float* A, const float* B,
                                     v8f acc, int col, int half) {
    v2f a0, a1, b0, b1;
#pragma unroll
    for (int r = 0; r < 2; ++r) {
        int kk0 = r + 2 * half;        // 0..3
        int kk1 = 4 + r + 2 * half;    // 4..7 (6,7 read zero padding)
        a0[r] = A[col * 16 + kk0];
        a1[r] = A[col * 16 + kk1];
        b0[r] = BT ? B[col * 16 + kk0] : B[kk0 * 16 + col];
        b1[r] = BT ? B[col * 16 + kk1] : B[kk1 * 16 + col];
    }
    acc = __builtin_amdgcn_wmma_f32_16x16x4_f32(false, a0, false, b0,
                                                (short)0, acc, false, false);
    acc = __builtin_amdgcn_wmma_f32_16x16x4_f32(false, a1, false, b1,
                                                (short)0, acc, false, false);
    return acc;
}

__device__ __forceinline__ void store6p(float* dst, v8f d, int col, int half) {
#pragma unroll
    for (int r = 0; r < 8; ++r) dst[(r + 8 * half) * 16 + col] = d[r];
}

__device__ __forceinline__ v8f load6p(const float* src, int col, int half) {
    v8f c;
#pragma unroll
    for (int r = 0; r < 8; ++r) c[r] = src[(r + 8 * half) * 16 + col];
    return c;
}

// uniform 3x3: S = H P H^T + R from padded tile -> store detS, Sinv to ws
__device__ __forceinline__ void evalS_store(const float* Pb, float R0, float R1,
                                            float R2, float* ws, int t) {
    float s00 = Pb[0] + R0,  s01 = Pb[1],       s02 = Pb[4];
    float s10 = Pb[16],      s11 = Pb[17] + R1, s12 = Pb[20];
    float s20 = Pb[64],      s21 = Pb[65],      s22 = Pb[68] + R2;
    float c00 = s11 * s22 - s12 * s21;
    float c01 = s12 * s20 - s10 * s22;
    float c02 = s10 * s21 - s11 * s20;
    float det = s00 * c00 + s01 * c01 + s02 * c02;
    float id  = 1.f / det;
    float* Sv = ws + SINV_OFF + t * 9;
    Sv[0] = c00 * id; Sv[1] = (s02 * s21 - s01 * s22) * id; Sv[2] = (s01 * s12 - s02 * s11) * id;
    Sv[3] = c01 * id; Sv[4] = (s00 * s22 - s02 * s20) * id; Sv[5] = (s02 * s10 - s00 * s12) * id;
    Sv[6] = c02 * id; Sv[7] = (s01 * s20 - s00 * s21) * id; Sv[8] = (s00 * s11 - s01 * s10) * id;
    ws[DETS_OFF + t] = det;
}

// ---------------------------------------------------------------------------
// Kernel 1: shared covariance/gain recursion, single wave, WMMA-driven.
// ---------------------------------------------------------------------------
__global__ void __launch_bounds__(32)
cov_kernel(const float* __restrict__ dyna, const float* __restrict__ cov,
           float* __restrict__ ws) {
    __shared__ float Pf_hist[TSTEPS * 36];                      // 36 KB compact
    // padded 16x16 tiles (zero outside 6x6)
    __shared__ float Fb[256], Pcur[256], T1b[256], Ppb[256], Invb[256];
    __shared__ float Cgb[256], Db[256], Psc[256], T3b[256], Pfp[256];
    __shared__ float Sinvb[9], Kb[18], Aug[72], fcol[6];

    const int lane = threadIdx.x;
    const int col  = lane & 15;
    const int half = lane >> 4;

    const float friction = dyna[0], damping = dyna[1];
    const float R0 = cov[0], R1 = cov[1], R2 = cov[2];
    const float q0 = cov[3], q1 = cov[4], q2 = cov[5], q3 = cov[6];

    // zero all padded tiles once
    for (int e = lane; e < 256; e += 32) {
        Fb[e] = 0.f; Pcur[e] = 0.f; T1b[e] = 0.f; Ppb[e] = 0.f; Invb[e] = 0.f;
        Cgb[e] = 0.f; Db[e] = 0.f; Psc[e] = 0.f; T3b[e] = 0.f; Pfp[e] = 0.f;
    }
    __syncthreads();

    for (int e = lane; e < 36; e += 32) {
        int i = e / 6, j = e % 6;
        float v = (i == j) ? 1.f : 0.f;
        if (i == 0 && j == 2) v = DT_;
        if (i == 1 && j == 3) v = DT_;
        if (i == 4 && j == 5) v = DT_;
        if (i == 2 && j == 2) v = 1.f - DT_ * damping;
        if (i == 3 && j == 3) v = 1.f - DT_ * damping;
        if (i == 5 && j == 5) v = 1.f - DT_ * friction;
        Fb[i * 16 + j]   = v;
        Pcur[i * 16 + j] = (i == j) ? P0S : 0.f;
    }
    __syncthreads();

    // Q (diagonal) in WMMA C/D layout, reused every step
    v8f qc;
#pragma unroll
    for (int r = 0; r < 8; ++r) qc[r] = 0.f;
    if (half == 0) {
        if (col == 0) qc[0] = q0;
        if (col == 1) qc[1] = q0;
        if (col == 2) qc[2] = q1;
        if (col == 3) qc[3] = q1;
        if (col == 4) qc[4] = q2;
        if (col == 5) qc[5] = q3;
    }
    v8f z8;
#pragma unroll
    for (int r = 0; r < 8; ++r) z8[r] = 0.f;

    // ---------------- forward Riccati: Pp = F P F^T + Q ; K ; Pf -----------
    for (int t = 0; t < TSTEPS; ++t) {
        v8f d = mul6p<false>(Fb, Pcur, z8, col, half);          // T1 = F @ P
        store6p(T1b, d, col, half);
        __syncthreads();
        d = mul6p<true>(T1b, Fb, qc, col, half);                // Pp = T1 @ F^T + Q
        store6p(Ppb, d, col, half);
        __syncthreads();

        if (lane == 0) {                                        // Sinv (3x3)
            float s00 = Ppb[0] + R0,  s01 = Ppb[1],       s02 = Ppb[4];
            float s10 = Ppb[16],      s11 = Ppb[17] + R1, s12 = Ppb[20];
            float s20 = Ppb[64],      s21 = Ppb[65],      s22 = Ppb[68] + R2;
            float c00 = s11 * s22 - s12 * s21;
            float c01 = s12 * s20 - s10 * s22;
            float c02 = s10 * s21 - s11 * s20;
            float id  = 1.f / (s00 * c00 + s01 * c01 + s02 * c02);
            Sinvb[0] = c00 * id; Sinvb[1] = (s02 * s21 - s01 * s22) * id; Sinvb[2] = (s01 * s12 - s02 * s11) * id;
            Sinvb[3] = c01 * id; Sinvb[4] = (s00 * s22 - s02 * s20) * id; Sinvb[5] = (s02 * s10 - s00 * s12) * id;
            Sinvb[6] = c02 * id; Sinvb[7] = (s01 * s20 - s00 * s21) * id; Sinvb[8] = (s00 * s11 - s01 * s10) * id;
        }
        __syncthreads();

        if (lane < 18) {                                        // K = Pp H^T Sinv
            int i = lane / 3, m = lane % 3;
            float kv = Ppb[i * 16 + 0] * Sinvb[m] +
                       Ppb[i * 16 + 1] * Sinvb[3 + m] +
                       Ppb[i * 16 + 4] * Sinvb[6 + m];
            Kb[lane] = kv;
            ws[K_OFF + t * 18 + lane] = kv;
        }
        __syncthreads();

        for (int e = lane; e < 36; e += 32) {                   // Pf = Pp - K(H Pp)
            int i = e / 6, j = e % 6;
            float v = Ppb[i * 16 + j] - Kb[i * 3 + 0] * Ppb[j] -
                                        Kb[i * 3 + 1] * Ppb[16 + j] -
                                        Kb[i * 3 + 2] * Ppb[64 + j];
            Pf_hist[t * 36 + e] = v;
            Pcur[i * 16 + j] = v;
        }
        __syncthreads();
    }

    // ---------------- backward RTS covariance / gains ----------------------
    for (int e = lane; e < 36; e += 32)
        Psc[(e / 6) * 16 + (e % 6)] = Pf_hist[255 * 36 + e];
    __syncthreads();
    if (lane == 0) evalS_store(Psc, R0, R1, R2, ws, 255);
    __syncthreads();

    for (int t = 254; t >= 0; --t) {
        // stage Pf_t into a padded tile
        for (int e = lane; e < 36; e += 32)
            Pfp[(e / 6) * 16 + (e % 6)] = Pf_hist[t * 36 + e];
        __syncthreads();

        v8f d = mul6p<false>(Fb, Pfp, z8, col, half);           // pp = F Pf F^T + Q
        store6p(T1b, d, col, half);
        __syncthreads();
        d = mul6p<true>(T1b, Fb, qc, col, half);
        store6p(Ppb, d, col, half);
        __syncthreads();

        // Gauss-Jordan 6x6 inverse of pp (lane-parallel in LDS)
        for (int e = lane; e < 72; e += 32) {
            int i = e / 12, j = e % 12;
            Aug[e] = (j < 6) ? Ppb[i * 16 + j] : ((j - 6 == i) ? 1.f : 0.f);
        }
        __syncthreads();
        for (int k = 0; k < 6; ++k) {
            if (lane < 12) {
                float pv = Aug[k * 12 + k];
                Aug[k * 12 + lane] = Aug[k * 12 + lane] / pv;
            }
            __syncthreads();
            if (lane < 6) fcol[lane] = (lane == k) ? 0.f : Aug[lane * 12 + k];
            __syncthreads();
            for (int e = lane; e < 72; e += 32) {
                int i = e / 12, j = e % 12;
                Aug[e] -= fcol[i] * Aug[k * 12 + j];
            }
            __syncthreads();
        }
        for (int e = lane; e < 36; e += 32) {
            int i = e / 6, j = e % 6;
            Invb[i * 16 + j] = Aug[i * 12 + 6 + j];
        }
        __syncthreads();

        d = mul6p<true>(Pfp, Fb, z8, col, half);                // T2 = Pf @ F^T
        store6p(T1b, d, col, half);
        __syncthreads();
        d = mul6p<false>(T1b, Invb, z8, col, half);             // C = T2 @ pp^-1
        store6p(Cgb, d, col, half);
        __syncthreads();
        for (int e = lane; e < 36; e += 32)                     // C -> workspace
            ws[C_OFF + t * 36 + e] = Cgb[(e / 6) * 16 + (e % 6)];

        for (int e = lane; e < 36; e += 32) {                   // D = ps - pp
            int i = e / 6, j = e % 6;
            Db[i * 16 + j] = Psc[i * 16 + j] - Ppb[i * 16 + j];
        }
        __syncthreads();
        d = mul6p<false>(Cgb, Db, z8, col, half);               // T3 = C @ (ps-pp)
        store6p(T3b, d, col, half);
        __syncthreads();
        v8f accPf = load6p(Pfp, col, half);
        d = mul6p<true>(T3b, Cgb, accPf, col, half);            // ps' = T3 C^T + Pf
        store6p(Psc, d, col, half);
        __syncthreads();
        if (lane == 0) evalS_store(Psc, R0, R1, R2, ws, t);
        __syncthreads();
    }
}

// ---------------------------------------------------------------------------
// Kernel 2: per-trajectory forward filtered states (sparse F, K_t shared).
// ---------------------------------------------------------------------------
__global__ void __launch_bounds__(256)
state_forward(const float* __restrict__ obs, const float* __restrict__ x0s,
              const float* __restrict__ dyna, float* __restrict__ ws) {
    int b = blockIdx.x * blockDim.x + threadIdx.x;
    const float av = 1.f - DT_ * dyna[1];   // damping
    const float aw = 1.f - DT_ * dyna[0];   // friction
    float x0 = x0s[b * 6 + 0], x1 = x0s[b * 6 + 1], x2 = x0s[b * 6 + 2];
    float x3 = x0s[b * 6 + 3], x4 = x0s[b * 6 + 4], x5 = x0s[b * 6 + 5];
    const float* K  = ws + K_OFF;
    float*       xf = ws + XF_OFF;
    const float* z  = obs + (size_t)b * TSTEPS * 3;

    for (int t = 0; t < TSTEPS; ++t) {
        float xp0 = x0 + DT_ * x2, xp1 = x1 + DT_ * x3;
        float xp2 = av * x2,       xp3 = av * x3;
        float xp4 = x4 + DT_ * x5, xp5 = aw * x5;
        float y0 = z[t * 3 + 0] - xp0;
        float y1 = z[t * 3 + 1] - xp1;
        float y2 = wrapf(z[t * 3 + 2] - xp4);
        const float* Kt = K + t * 18;
        x0 = xp0 + Kt[0]  * y0 + Kt[1]  * y1 + Kt[2]  * y2;
        x1 = xp1 + Kt[3]  * y0 + Kt[4]  * y1 + Kt[5]  * y2;
        x2 = xp2 + Kt[6]  * y0 + Kt[7]  * y1 + Kt[8]  * y2;
        x3 = xp3 + Kt[9]  * y0 + Kt[10] * y1 + Kt[11] * y2;
        x4 = xp4 + Kt[12] * y0 + Kt[13] * y1 + Kt[14] * y2;
        x5 = xp5 + Kt[15] * y0 + Kt[16] * y1 + Kt[17] * y2;
        size_t o = ((size_t)t * BATCH + b) * 6;
        xf[o + 0] = x0; xf[o + 1] = x1; xf[o + 2] = x2;
        xf[o + 3] = x3; xf[o + 4] = x4; xf[o + 5] = x5;
    }
}

// ---------------------------------------------------------------------------
// Kernel 3: backward state smoothing fused with NLL eval; block partial sums.
// ---------------------------------------------------------------------------
__global__ void __launch_bounds__(256)
state_backward(const float* __restrict__ obs, const float* __restrict__ dyna,
               float* __restrict__ ws) {
    int b = blockIdx.x * blockDim.x + threadIdx.x;
    const float av = 1.f - DT_ * dyna[1];
    const float aw = 1.f - DT_ * dyna[0];
    const float* C    = ws + C_OFF;
    const float* Sinv = ws + SINV_OFF;
    const float* dS   = ws + DETS_OFF;
    const float* xf   = ws + XF_OFF;
    const float* z    = obs + (size_t)b * TSTEPS * 3;

    size_t o = ((size_t)255 * BATCH + b) * 6;
    float s0 = xf[o + 0], s1 = xf[o + 1], s2 = xf[o + 2];
    float s3 = xf[o + 3], s4 = xf[o + 4], s5 = xf[o + 5];

    float loss = 0.f;
    {   // t = 255 eval
        const float* Sv = Sinv + 255 * 9;
        float y0 = z[255 * 3 + 0] - s0, y1 = z[255 * 3 + 1] - s1;
        float y2 = wrapf(z[255 * 3 + 2] - s4);
        loss += dS[255] +
                y0 * (Sv[0] * y0 + Sv[1] * y1 + Sv[2] * y2) +
                y1 * (Sv[3] * y0 + Sv[4] * y1 + Sv[5] * y2) +
                y2 * (Sv[6] * y0 + Sv[7] * y1 + Sv[8] * y2);
    }

    for (int t = 254; t >= 0; --t) {
        o = ((size_t)t * BATCH + b) * 6;
        float f0 = xf[o + 0], f1 = xf[o + 1], f2 = xf[o + 2];
        float f3 = xf[o + 3], f4 = xf[o + 4], f5 = xf[o + 5];
        float d0 = s0 - (f0 + DT_ * f2);
        float d1 = s1 - (f1 + DT_ * f3);
        float d2 = s2 - av * f2;
        float d3 = s3 - av * f3;
        float d4 = wrapf(s4 - (f4 + DT_ * f5));
        float d5 = s5 - aw * f5;
        const float* Ct = C + t * 36;
        s0 = f0 + Ct[0]  * d0 + Ct[1]  * d1 + Ct[2]  * d2 + Ct[3]  * d3 + Ct[4]  * d4 + Ct[5]  * d5;
        s1 = f1 + Ct[6]  * d0 + Ct[7]  * d1 + Ct[8]  * d2 + Ct[9]  * d3 + Ct[10] * d4 + Ct[11] * d5;
        s2 = f2 + Ct[12] * d0 + Ct[13] * d1 + Ct[14] * d2 + Ct[15] * d3 + Ct[16] * d4 + Ct[17] * d5;
        s3 = f3 + Ct[18] * d0 + Ct[19] * d1 + Ct[20] * d2 + Ct[21] * d3 + Ct[22] * d4 + Ct[23] * d5;
        s4 = f4 + Ct[24] * d0 + Ct[25] * d1 + Ct[26] * d2 + Ct[27] * d3 + Ct[28] * d4 + Ct[29] * d5;
        s5 = f5 + Ct[30] * d0 + Ct[31] * d1 + Ct[32] * d2 + Ct[33] * d3 + Ct[34] * d4 + Ct[35] * d5;

        const float* Sv = Sinv + t * 9;
        float y0 = z[t * 3 + 0] - s0, y1 = z[t * 3 + 1] - s1;
        float y2 = wrapf(z[t * 3 + 2] - s4);
        loss += dS[t] +
                y0 * (Sv[0] * y0 + Sv[1] * y1 + Sv[2] * y2) +
                y1 * (Sv[3] * y0 + Sv[4] * y1 + Sv[5] * y2) +
                y2 * (Sv[6] * y0 + Sv[7] * y1 + Sv[8] * y2);
    }

    __shared__ float red[256];
    red[threadIdx.x] = loss;
    __syncthreads();
    for (int s = 128; s > 0; s >>= 1) {
        if (threadIdx.x < s) red[threadIdx.x] += red[threadIdx.x + s];
        __syncthreads();
    }
    if (threadIdx.x == 0) ws[PART_OFF + blockIdx.x] = red[0];
}

__global__ void final_reduce(const float* __restrict__ ws, float* __restrict__ out) {
    float s = 0.f;
    for (int i = 0; i < BATCH / 256; ++i) s += ws[PART_OFF + i];
    out[0] = s * (1.0f / ((float)BATCH * (float)TSTEPS));
}

extern "C" void kernel_launch(void* const* d_in, const int* in_sizes, int n_in,
                              void* d_out, int out_size, void* d_ws, size_t ws_size,
                              hipStream_t stream) {
    const float* obs  = (const float*)d_in[0];   // [4096,256,3]
    const float* x0s  = (const float*)d_in[1];   // [4096,6]
    const float* dyna = (const float*)d_in[2];   // [4]
    const float* cov  = (const float*)d_in[3];   // [7]
    float* ws  = (float*)d_ws;
    float* out = (float*)d_out;

    cov_kernel    <<<1, 32, 0, stream>>>(dyna, cov, ws);
    state_forward <<<BATCH / 256, 256, 0, stream>>>(obs, x0s, dyna, ws);
    state_backward<<<BATCH / 256, 256, 0, stream>>>(obs, dyna, ws);
    final_reduce  <<<1, 1, 0, stream>>>(ws, out);
}